// MeshConv_76819785056399
// MI455X (gfx1250) — compile-verified
//
#include <hip/hip_runtime.h>

typedef __attribute__((ext_vector_type(2))) float v2f;
typedef __attribute__((ext_vector_type(8))) float v8f;

// ---------------------------------------------------------------------------
// GEMM:  T[b][n][o] = sum_c x[b][c][n] * W[c][o]  (+ bias[o] if HAS_BIAS)
// x: [4][64][nv], W: [64][64], T: [4][nv][64]
// One wave32 per (b, 16-row tile of n). Uses V_WMMA_F32_16X16X4_F32 (exact fp32).
//   A tile 16x4 (MxK): vgpr0 lanes0-15 = K0, lanes16-31 = K2; vgpr1 = K1/K3; M = lane%16
//   B tile 4x16 (KxN): same K striping; N = lane%16
//   C/D 16x16: vgpr r, lanes0-15: M=r, lanes16-31: M=r+8; N = lane%16
// ---------------------------------------------------------------------------
template <bool HAS_BIAS>
__global__ __launch_bounds__(256)
void gemm_xw_kernel(const float* __restrict__ x, const float* __restrict__ W,
                    const float* __restrict__ bias, float* __restrict__ T,
                    int nv, int mtiles)
{
    int wave = (int)((blockIdx.x * blockDim.x + threadIdx.x) >> 5);
    int lane = (int)(threadIdx.x & 31);
    if (wave >= 4 * mtiles) return;          // uniform per-wave: EXEC stays all-ones
    int b  = wave / mtiles;
    int nb = (wave % mtiles) * 16;
    int half = lane >> 4;                    // selects K pair (0->K0/K1, 1->K2/K3)
    int lh   = lane & 15;                    // M (for A) / N (for B,C,D)

    const float* xb = x + (size_t)b * 64 * (size_t)nv;

    int n = nb + lh;
    if (n >= nv) n = nv - 1;                 // clamp loads; stores are masked below

    v8f acc0 = {0.f,0.f,0.f,0.f,0.f,0.f,0.f,0.f};
    v8f acc1 = acc0, acc2 = acc0, acc3 = acc0;

    for (int kk = 0; kk < 16; ++kk) {
        int c0 = kk * 4 + half * 2;
        v2f a;
        a.x = xb[(size_t)c0       * nv + n];
        a.y = xb[(size_t)(c0 + 1) * nv + n];

        v2f b0, b1, b2, b3;
        b0.x = W[c0 * 64 +  0 + lh];  b0.y = W[(c0 + 1) * 64 +  0 + lh];
        b1.x = W[c0 * 64 + 16 + lh];  b1.y = W[(c0 + 1) * 64 + 16 + lh];
        b2.x = W[c0 * 64 + 32 + lh];  b2.y = W[(c0 + 1) * 64 + 32 + lh];
        b3.x = W[c0 * 64 + 48 + lh];  b3.y = W[(c0 + 1) * 64 + 48 + lh];

        acc0 = __builtin_amdgcn_wmma_f32_16x16x4_f32(false, a, false, b0, (short)0, acc0, false, false);
        acc1 = __builtin_amdgcn_wmma_f32_16x16x4_f32(false, a, false, b1, (short)0, acc1, false, false);
        acc2 = __builtin_amdgcn_wmma_f32_16x16x4_f32(false, a, false, b2, (short)0, acc2, false, false);
        acc3 = __builtin_amdgcn_wmma_f32_16x16x4_f32(false, a, false, b3, (short)0, acc3, false, false);
    }

    float bv0 = 0.f, bv1 = 0.f, bv2 = 0.f, bv3 = 0.f;
    if (HAS_BIAS) {
        bv0 = bias[ 0 + lh];
        bv1 = bias[16 + lh];
        bv2 = bias[32 + lh];
        bv3 = bias[48 + lh];
    }
    // Store: T[(b*nv + n)*64 + o]; lanes 0-15 give o contiguous (64B segments).
#pragma unroll
    for (int r = 0; r < 8; ++r) {
        int nn = nb + r + 8 * half;
        if (nn < nv) {
            float* row = T + ((size_t)b * nv + nn) * 64;
            row[ 0 + lh] = acc0[r] + bv0;
            row[16 + lh] = acc1[r] + bv1;
            row[32 + lh] = acc2[r] + bv2;
            row[48 + lh] = acc3[r] + bv3;
        }
    }
}

// ---------------------------------------------------------------------------
// Counting-sort CSR build (turns 325M scatter atomics into ~2.5M + sorted gather)
// ---------------------------------------------------------------------------
__global__ void zero_u32_kernel(unsigned* __restrict__ p, int n)
{
    int i = (int)(blockIdx.x * blockDim.x + threadIdx.x);
    if (i < n) p[i] = 0u;
}

__global__ void hist_kernel(const int* __restrict__ row, int nnz, unsigned* __restrict__ cnt)
{
    int i = (int)(blockIdx.x * blockDim.x + threadIdx.x);
    if (i < nnz) atomicAdd(&cnt[row[i]], 1u);
}

// Single-block exclusive scan, in place: rowptr holds counts on entry, offsets on exit.
// Also copies the offsets into cursor[] for the atomic scatter pass.
__global__ __launch_bounds__(1024)
void scan_kernel(unsigned* __restrict__ rowptr, unsigned* __restrict__ cursor, int nv)
{
    __shared__ unsigned sdata[1024];
    __shared__ unsigned carry;
    int tid = (int)threadIdx.x;
    if (tid == 0) carry = 0u;
    __syncthreads();
    for (int base = 0; base < nv; base += 1024) {
        int idx = base + tid;
        unsigned v = (idx < nv) ? rowptr[idx] : 0u;
        sdata[tid] = v;
        __syncthreads();
        for (int off = 1; off < 1024; off <<= 1) {
            unsigned t = (tid >= off) ? sdata[tid - off] : 0u;
            __syncthreads();
            sdata[tid] += t;
            __syncthreads();
        }
        unsigned excl = sdata[tid] - v + carry;
        if (idx < nv) { rowptr[idx] = excl; cursor[idx] = excl; }
        __syncthreads();                       // all lanes read carry before update
        if (tid == 1023) carry += sdata[1023];
        __syncthreads();
    }
    if (tid == 0) { rowptr[nv] = carry; cursor[nv] = carry; }
}

__global__ void csr_scatter_kernel(const int* __restrict__ row, const int* __restrict__ col,
                                   const float* __restrict__ val, int nnz,
                                   unsigned* __restrict__ cursor,
                                   int* __restrict__ scol, float* __restrict__ sval)
{
    int i = (int)(blockIdx.x * blockDim.x + threadIdx.x);
    if (i < nnz) {
        unsigned p = atomicAdd(&cursor[row[i]], 1u);
        scol[p] = col[i];
        sval[p] = val[i];
    }
}

// ---------------------------------------------------------------------------
// CSR apply: acc[b][n][o] += sum_{j in row n} sval[j] * Z[b][scol[j]][o]
// One wave32 per vertex n handling ALL 4 batches (CSR row read once, not 4x).
// Lane owns channels {lane, lane+32}; 8 accumulators/lane. Atomic-free.
// Next column index is software-pipelined and its Z rows prefetched
// (global_prefetch_b8) to hide the random-gather L2 hit latency.
// ---------------------------------------------------------------------------
__global__ __launch_bounds__(256)
void gather_kernel(const unsigned* __restrict__ rowptr, const int* __restrict__ scol,
                   const float* __restrict__ sval, const float* __restrict__ Z,
                   float* __restrict__ acc, int nv)
{
    int wave = (int)((blockIdx.x * blockDim.x + threadIdx.x) >> 5);
    int lane = (int)(threadIdx.x & 31);
    if (wave >= nv) return;
    int n = wave;
    unsigned s = rowptr[n], e = rowptr[n + 1];

    float a0b0 = 0.f, a1b0 = 0.f, a0b1 = 0.f, a1b1 = 0.f;
    float a0b2 = 0.f, a1b2 = 0.f, a0b3 = 0.f, a1b3 = 0.f;

    if (s < e) {
        int c = scol[s];
        for (unsigned j = s; j < e; ++j) {
            unsigned jn = (j + 1 < e) ? (j + 1) : (e - 1);
            int cn = scol[jn];                 // next column (pipelined)
            const float* zp = Z + (size_t)cn * 64;
            __builtin_prefetch(zp,                              0, 1);
            __builtin_prefetch(zp + (size_t)1 * nv * 64,        0, 1);
            __builtin_prefetch(zp + (size_t)2 * nv * 64,        0, 1);
            __builtin_prefetch(zp + (size_t)3 * nv * 64,        0, 1);

            float v = sval[j];
            const float* zr = Z + (size_t)c * 64;
            a0b0 += v * zr[lane];
            a1b0 += v * zr[lane + 32];
            zr += (size_t)nv * 64;
            a0b1 += v * zr[lane];
            a1b1 += v * zr[lane + 32];
            zr += (size_t)nv * 64;
            a0b2 += v * zr[lane];
            a1b2 += v * zr[lane + 32];
            zr += (size_t)nv * 64;
            a0b3 += v * zr[lane];
            a1b3 += v * zr[lane + 32];
            c = cn;
        }
    }

    float* ar = acc + (size_t)n * 64;
    ar[lane]      += a0b0;  ar[lane + 32] += a1b0;
    ar += (size_t)nv * 64;
    ar[lane]      += a0b1;  ar[lane + 32] += a1b1;
    ar += (size_t)nv * 64;
    ar[lane]      += a0b2;  ar[lane + 32] += a1b2;
    ar += (size_t)nv * 64;
    ar[lane]      += a0b3;  ar[lane + 32] += a1b3;
}

// ---------------------------------------------------------------------------
// Final transpose: out[b][o][n] = acc[b][n][o]  (32x32 LDS tiles, coalesced)
// ---------------------------------------------------------------------------
__global__ __launch_bounds__(256)
void transpose_kernel(const float* __restrict__ acc, float* __restrict__ out, int nv)
{
    __shared__ float tile[32][33];
    int b     = (int)blockIdx.z;
    int obase = (int)blockIdx.y * 32;
    int nbase = (int)blockIdx.x * 32;
#pragma unroll
    for (int i = 0; i < 4; ++i) {
        int n = nbase + (int)threadIdx.y + i * 8;
        int o = obase + (int)threadIdx.x;
        float v = (n < nv) ? acc[((size_t)b * nv + n) * 64 + o] : 0.f;
        tile[threadIdx.y + i * 8][threadIdx.x] = v;
    }
    __syncthreads();
#pragma unroll
    for (int i = 0; i < 4; ++i) {
        int o = obase + (int)threadIdx.y + i * 8;
        int n = nbase + (int)threadIdx.x;
        if (n < nv) out[((size_t)b * 64 + o) * nv + n] = tile[threadIdx.x][threadIdx.y + i * 8];
    }
}

// ---------------------------------------------------------------------------
extern "C" void kernel_launch(void* const* d_in, const int* in_sizes, int n_in,
                              void* d_out, int out_size, void* d_ws, size_t ws_size,
                              hipStream_t stream)
{
    const float* x      = (const float*)d_in[0];
    const float* coeffs = (const float*)d_in[10];
    const float* bias   = (const float*)d_in[11];

    const int nv    = in_sizes[0] / (4 * 64);     // x is [4][64][nv]
    const int nnz_g = in_sizes[4];                // max nnz among ops (12*nv)

    // Workspace carve-out (~88 MB total)
    const size_t SZ = (size_t)4 * nv * 64 * sizeof(float);
    auto align256 = [](size_t v) { return (v + 255) & ~(size_t)255; };
    char* p = (char*)d_ws;
    float*    acc    = (float*)p;    p += align256(SZ);
    float*    Z      = (float*)p;    p += align256(SZ);
    unsigned* rowptr = (unsigned*)p; p += align256((size_t)(nv + 1) * 4);
    unsigned* cursor = (unsigned*)p; p += align256((size_t)(nv + 1) * 4);
    int*      scol   = (int*)p;      p += align256((size_t)nnz_g * 4);
    float*    sval   = (float*)p;    /* end */

    const int mtiles     = (nv + 15) / 16;
    const int gemm_blk   = (4 * mtiles + 7) / 8;      // 8 waves/block
    const int gather_blk = (nv + 7) / 8;              // 1 wave per vertex

    // Identity branch + bias -> acc  (W0 = coeffs[0])
    gemm_xw_kernel<true><<<gemm_blk, 256, 0, stream>>>(x, coeffs, bias, acc, nv, mtiles);

    // Sparse branches: (row,col,val) input slots and coeffs index per op
    const int rix[3] = {1, 4, 7};
    for (int op = 0; op < 3; ++op) {
        const int*   row = (const int*)  d_in[rix[op] + 0];
        const int*   col = (const int*)  d_in[rix[op] + 1];
        const float* val = (const float*)d_in[rix[op] + 2];
        const int    nnz = in_sizes[rix[op]];
        const float* Wk  = coeffs + (size_t)(op + 1) * 64 * 64;

        zero_u32_kernel<<<(nv + 1 + 255) / 256, 256, 0, stream>>>(rowptr, nv + 1);
        hist_kernel<<<(nnz + 255) / 256, 256, 0, stream>>>(row, nnz, rowptr);
        scan_kernel<<<1, 1024, 0, stream>>>(rowptr, cursor, nv);
        csr_scatter_kernel<<<(nnz + 255) / 256, 256, 0, stream>>>(row, col, val, nnz,
                                                                  cursor, scol, sval);
        gemm_xw_kernel<false><<<gemm_blk, 256, 0, stream>>>(x, Wk, nullptr, Z, nv, mtiles);
        gather_kernel<<<gather_blk, 256, 0, stream>>>(rowptr, scol, sval, Z, acc, nv);
    }

    // acc[b][n][o] -> out[b][o][n]
    dim3 tgrid((nv + 31) / 32, 2, 4);
    dim3 tblk(32, 8);
    transpose_kernel<<<tgrid, tblk, 0, stream>>>(acc, (float*)d_out, nv);
}